// GATGraphRegressor_90993177133760
// MI455X (gfx1250) — compile-verified
//
#include <hip/hip_runtime.h>
#include <hip/hip_bf16.h>

typedef __attribute__((ext_vector_type(2))) float v2f;
typedef __attribute__((ext_vector_type(8))) float v8f;

#define DH 128
#define NEG_SLOPE 0.2f

// ---------------------------------------------------------------------------
// GEMM: H[n_nodes,128] = X[n_nodes,128] @ W[128,128] using V_WMMA_F32_16X16X4_F32
// One block (256 thr = 8 waves) per 16-row strip; wave w owns cols [16w,16w+16).
// X tile staged in LDS; W read through L0/L2 (64 KB, fully cache-resident).
// A layout (16x4 f32): lane<16 -> {K=k,k+1}, lane>=16 -> {K=k+2,k+3}, M=lane&15
// B layout (4x16 f32): mirrored, N=lane&15
// C/D: VGPR i, lane<16 -> M=i, lane>=16 -> M=i+8, N=lane&15
// ---------------------------------------------------------------------------
__global__ void __launch_bounds__(256) k_gemm_wmma(
    const float* __restrict__ X, const float* __restrict__ W,
    float* __restrict__ H, int n_nodes)
{
  __shared__ float sX[16 * DH];                      // 8 KB tile
  const int row0 = blockIdx.x * 16;
  const int tid  = threadIdx.x;

  float4* sX4 = (float4*)sX;
  const float4* Xg = (const float4*)(X + (size_t)row0 * DH);
  for (int i = tid; i < 16 * DH / 4; i += 256) {     // 512 float4, coalesced
    int r = i >> 5;                                  // 32 float4 per row
    float4 v = make_float4(0.f, 0.f, 0.f, 0.f);
    if (row0 + r < n_nodes) v = Xg[i];
    sX4[i] = v;
  }
  __syncthreads();

  const int wave = tid >> 5;
  const int lane = tid & 31;
  const int col0 = wave * 16;
  const int mn   = lane & 15;                        // A row / B col
  const int kh   = (lane >> 4) * 2;                  // K sub-offset 0 or 2

  v8f c = {};
  for (int k = 0; k < DH; k += 4) {
    v2f a, b;
    a[0] = sX[mn * DH + k + kh];
    a[1] = sX[mn * DH + k + kh + 1];
    b[0] = W[(k + kh)     * DH + col0 + mn];
    b[1] = W[(k + kh + 1) * DH + col0 + mn];
    c = __builtin_amdgcn_wmma_f32_16x16x4_f32(false, a, false, b,
                                              (short)0, c, false, false);
  }

  const int mbase = (lane < 16) ? 0 : 8;
  #pragma unroll
  for (int i = 0; i < 8; ++i) {
    int r = row0 + mbase + i;
    if (r < n_nodes) H[(size_t)r * DH + col0 + mn] = c[i];
  }
}

// ---------------------------------------------------------------------------
// alpha_s[n] = h[n,:] . a_src ; alpha_d[n] = h[n,:] . a_dst   (wave per node)
// ---------------------------------------------------------------------------
__global__ void __launch_bounds__(256) k_alpha(
    const float* __restrict__ H, const float* __restrict__ a_src,
    const float* __restrict__ a_dst, float* __restrict__ as_,
    float* __restrict__ ad_, int n_nodes)
{
  int node = (int)((blockIdx.x * blockDim.x + threadIdx.x) >> 5);
  int lane = threadIdx.x & 31;
  if (node >= n_nodes) return;
  const float* hr = H + (size_t)node * DH;
  float s = 0.f, d = 0.f;
  #pragma unroll
  for (int c = lane; c < DH; c += 32) {
    float hv = hr[c];
    s += hv * a_src[c];
    d += hv * a_dst[c];
  }
  #pragma unroll
  for (int off = 16; off > 0; off >>= 1) {
    s += __shfl_xor(s, off, 32);
    d += __shfl_xor(d, off, 32);
  }
  if (lane == 0) { as_[node] = s; ad_[node] = d; }
}

__global__ void k_zero_f32(float* __restrict__ p, long long n)
{
  long long i = (long long)blockIdx.x * blockDim.x + threadIdx.x;
  if (i < n) p[i] = 0.f;
}

// ordered-uint encoding so unsigned atomicMax == float max (handles negatives)
__device__ __forceinline__ unsigned f2ord(float f) {
  unsigned b = __float_as_uint(f);
  return (b & 0x80000000u) ? ~b : (b | 0x80000000u);
}
__device__ __forceinline__ float ord2f(unsigned u) {
  unsigned b = (u & 0x80000000u) ? (u & 0x7FFFFFFFu) : ~u;
  return __uint_as_float(b);
}

// pass 1: e = leaky_relu(as[src]+ad[dst]); segment max into mOrd[dst]
__global__ void k_edge_max(
    const int* __restrict__ ei, int NE, int n_nodes,
    const float* __restrict__ as_, const float* __restrict__ ad_,
    float* __restrict__ ebuf, unsigned* __restrict__ mOrd)
{
  int e = blockIdx.x * blockDim.x + threadIdx.x;
  int E = NE + n_nodes;
  if (e >= E) return;
  int s, d;
  if (e < NE) { s = ei[e]; d = ei[NE + e]; } else { s = d = e - NE; }
  float v = as_[s] + ad_[d];
  v = (v >= 0.f) ? v : NEG_SLOPE * v;
  ebuf[e] = v;
  atomicMax(&mOrd[d], f2ord(v));
}

// pass 2: ex = exp(e - m[dst]); denom[dst] += ex
__global__ void k_edge_exp(
    const int* __restrict__ ei, int NE, int n_nodes,
    float* __restrict__ ebuf, const unsigned* __restrict__ mOrd,
    float* __restrict__ denom)
{
  int e = blockIdx.x * blockDim.x + threadIdx.x;
  int E = NE + n_nodes;
  if (e >= E) return;
  int d = (e < NE) ? ei[NE + e] : (e - NE);
  float ex = __expf(ebuf[e] - ord2f(mOrd[d]));
  ebuf[e] = ex;
  atomicAdd(&denom[d], ex);
}

// pass 3: out[dst,:] += h[src,:] * (ex / (denom[dst]+1e-16))   (wave per edge)
__global__ void __launch_bounds__(256) k_edge_agg(
    const int* __restrict__ ei, int NE, int n_nodes,
    const float* __restrict__ H, const float* __restrict__ ebuf,
    const float* __restrict__ denom, float* __restrict__ OUT)
{
  int e    = blockIdx.x * 8 + (threadIdx.x >> 5);
  int lane = threadIdx.x & 31;
  int E = NE + n_nodes;
  if (e >= E) return;
  int s, d;
  if (e < NE) { s = ei[e]; d = ei[NE + e]; } else { s = d = e - NE; }
  float coef = ebuf[e] / (denom[d] + 1e-16f);
  int c0 = lane * 4;
  float4 hv = *(const float4*)(H + (size_t)s * DH + c0);
  float* o = OUT + (size_t)d * DH + c0;
  atomicAdd(o + 0, hv.x * coef);
  atomicAdd(o + 1, hv.y * coef);
  atomicAdd(o + 2, hv.z * coef);
  atomicAdd(o + 3, hv.w * coef);
}

__global__ void k_bias_relu(float* __restrict__ A, const float* __restrict__ b,
                            long long n_elems)
{
  long long i = (long long)blockIdx.x * blockDim.x + threadIdx.x;
  if (i >= n_elems) return;
  float v = A[i] + b[i & (DH - 1)];
  A[i] = v > 0.f ? v : 0.f;
}

__global__ void k_pool(const float* __restrict__ H, const int* __restrict__ batch,
                       float* __restrict__ pooled, float* __restrict__ counts,
                       int n_nodes)
{
  long long i = (long long)blockIdx.x * blockDim.x + threadIdx.x;
  if (i >= (long long)n_nodes * DH) return;
  int node = (int)(i >> 7);
  int c    = (int)(i & (DH - 1));
  int g    = batch[node];
  atomicAdd(&pooled[(size_t)g * DH + c], H[i]);
  if (c == 0) atomicAdd(&counts[g], 1.0f);
}

// wave per graph: out[g] = (pooled[g,:]/max(cnt,1)) . fcW + fcb
__global__ void __launch_bounds__(256) k_final(
    const float* __restrict__ pooled, const float* __restrict__ counts,
    const float* __restrict__ fcW, const float* __restrict__ fcb,
    float* __restrict__ out, int n_graphs)
{
  int g    = blockIdx.x * 8 + (threadIdx.x >> 5);
  int lane = threadIdx.x & 31;
  if (g >= n_graphs) return;
  float inv = 1.0f / fmaxf(counts[g], 1.0f);
  float s = 0.f;
  #pragma unroll
  for (int c = lane; c < DH; c += 32) s += pooled[(size_t)g * DH + c] * fcW[c];
  #pragma unroll
  for (int off = 16; off > 0; off >>= 1) s += __shfl_xor(s, off, 32);
  if (lane == 0) out[g] = s * inv + fcb[0];
}

// ---------------------------------------------------------------------------
extern "C" void kernel_launch(void* const* d_in, const int* in_sizes, int n_in,
                              void* d_out, int out_size, void* d_ws, size_t ws_size,
                              hipStream_t stream)
{
  const float* x     = (const float*)d_in[0];
  const int*   ei    = (const int*)d_in[1];
  const int*   batch = (const int*)d_in[2];
  const float* fcW   = (const float*)d_in[19];
  const float* fcb   = (const float*)d_in[20];

  const int n_nodes  = in_sizes[2];
  const int NE       = in_sizes[1] / 2;
  const int E        = NE + n_nodes;
  const int n_graphs = out_size;   // [N_GRAPHS, 1] output

  // workspace carve-up (~56 MB for the reference sizes)
  char* ws = (char*)d_ws;
  size_t off = 0;
  auto carve = [&](size_t bytes) -> void* {
    void* p = ws + off;
    off += (bytes + 255) & ~(size_t)255;
    return p;
  };
  float*    bufH   = (float*)   carve((size_t)n_nodes * DH * sizeof(float));
  float*    bufX   = (float*)   carve((size_t)n_nodes * DH * sizeof(float));
  float*    as_    = (float*)   carve((size_t)n_nodes * sizeof(float));
  float*    ad_    = (float*)   carve((size_t)n_nodes * sizeof(float));
  unsigned* mOrd   = (unsigned*)carve((size_t)n_nodes * sizeof(unsigned));
  float*    denom  = (float*)   carve((size_t)n_nodes * sizeof(float));
  float*    ebuf   = (float*)   carve((size_t)E * sizeof(float));
  float*    pooled = (float*)   carve((size_t)n_graphs * DH * sizeof(float));
  float*    counts = (float*)   carve((size_t)n_graphs * sizeof(float));

  const long long nel = (long long)n_nodes * DH;
  const int blkE  = (E + 255) / 256;
  const int blkEw = (E + 7) / 8;
  const int blkN  = (int)((nel + 255) / 256);

  for (int layer = 0; layer < 4; ++layer) {
    const float* W  = (const float*)d_in[3 + 4 * layer];
    const float* av = (const float*)d_in[4 + 4 * layer];
    const float* aw = (const float*)d_in[5 + 4 * layer];
    const float* b  = (const float*)d_in[6 + 4 * layer];
    const float* in = (layer == 0) ? x : bufX;

    k_gemm_wmma<<<(n_nodes + 15) / 16, 256, 0, stream>>>(in, W, bufH, n_nodes);
    k_alpha<<<(n_nodes + 7) / 8, 256, 0, stream>>>(bufH, av, aw, as_, ad_, n_nodes);

    k_zero_f32<<<(n_nodes + 255) / 256, 256, 0, stream>>>((float*)mOrd, n_nodes);
    k_zero_f32<<<(n_nodes + 255) / 256, 256, 0, stream>>>(denom, n_nodes);
    k_zero_f32<<<blkN, 256, 0, stream>>>(bufX, nel);

    k_edge_max<<<blkE, 256, 0, stream>>>(ei, NE, n_nodes, as_, ad_, ebuf, mOrd);
    k_edge_exp<<<blkE, 256, 0, stream>>>(ei, NE, n_nodes, ebuf, mOrd, denom);
    k_edge_agg<<<blkEw, 256, 0, stream>>>(ei, NE, n_nodes, bufH, ebuf, denom, bufX);

    k_bias_relu<<<blkN, 256, 0, stream>>>(bufX, b, nel);
  }

  k_zero_f32<<<(n_graphs * DH + 255) / 256, 256, 0, stream>>>(pooled,
                                                             (long long)n_graphs * DH);
  k_zero_f32<<<1, 256, 0, stream>>>(counts, n_graphs);
  k_pool<<<blkN, 256, 0, stream>>>(bufX, batch, pooled, counts, n_nodes);
  k_final<<<(n_graphs + 7) / 8, 256, 0, stream>>>(pooled, counts, fcW, fcb,
                                                  (float*)d_out, n_graphs);
}